// GCNModel_6665789243503
// MI455X (gfx1250) — compile-verified
//
#include <hip/hip_runtime.h>
#include <math.h>

typedef __attribute__((ext_vector_type(2))) float v2f;
typedef __attribute__((ext_vector_type(4))) float v4f;
typedef __attribute__((ext_vector_type(8))) float v8f;

#define H128 128
#define LDSW_STRIDE 132   // padded row stride (floats): 16B-aligned rows, conflict-free half-wave reads

// ---------------------------------------------------------------------------
// Out[M x 128] = X[M x 128] @ W[128 x 128] (+ bias[n] if bias != nullptr)
// fp32 WMMA 16x16x4. 4 waves/block, wave = 16 rows x 128 cols (8 n-tiles).
// kt outer / nt inner: A loaded once per k-step, 8 independent WMMA chains.
// ---------------------------------------------------------------------------
__global__ __launch_bounds__(128)
void gemm128_wmma(const float* __restrict__ X, const float* __restrict__ W,
                  const float* __restrict__ bias, float* __restrict__ Out, int M)
{
    __shared__ float ldsW[H128 * LDSW_STRIDE];

    const int tid = threadIdx.x;

    // Stage W (row-major 128x128) into LDS with padded stride, float4 chunks.
    {
        const v4f* gsrc = (const v4f*)W;
        for (int i = tid; i < H128 * (H128 / 4); i += 128) {
            int k = i >> 5;         // row
            int q = i & 31;         // float4 within row
            *(v4f*)(ldsW + k * LDSW_STRIDE + q * 4) = gsrc[i];
        }
    }
    __syncthreads();

    const int wave = tid >> 5;
    const int lane = tid & 31;
    const int lo   = lane & 15;     // M index within tile (A/C), N index (B/C)
    const int hi   = lane >> 4;     // selects K pair (A/B), M+8 (C)

    const int mbase = blockIdx.x * 64 + wave * 16;
    // Clamp OOB rows to a valid address: their C rows are never stored, so the
    // values don't matter (each output row depends only on its own A row).
    int m = mbase + lo;
    if (m > M - 1) m = M - 1;
    const float* xrow = X + (size_t)m * H128;

    v8f c[8];
#pragma unroll
    for (int nt = 0; nt < 8; ++nt) c[nt] = (v8f){0.f,0.f,0.f,0.f,0.f,0.f,0.f,0.f};

#pragma unroll
    for (int kt = 0; kt < 32; ++kt) {
        const int k = kt * 4 + hi * 2;
        // A fragment: X[m][k], X[m][k+1]  (ISA 16x4 f32 layout)
        const v2f a = *(const v2f*)(xrow + k);
#pragma unroll
        for (int nt = 0; nt < 8; ++nt) {
            const int n = nt * 16 + lo;
            v2f b;
            b.x = ldsW[(k    ) * LDSW_STRIDE + n];   // B[k][n]
            b.y = ldsW[(k + 1) * LDSW_STRIDE + n];   // B[k+1][n]
            c[nt] = __builtin_amdgcn_wmma_f32_16x16x4_f32(
                        /*neg_a=*/false, a, /*neg_b=*/false, b,
                        /*c_mod=*/(short)0, c[nt], /*reuse_a=*/false, /*reuse_b=*/false);
        }
    }

#pragma unroll
    for (int nt = 0; nt < 8; ++nt) {
        const int n = nt * 16 + lo;
        const float bv = bias ? bias[n] : 0.0f;
#pragma unroll
        for (int v = 0; v < 8; ++v) {
            const int mr = mbase + v + hi * 8;       // C layout: VGPR v -> M = v (+8 for hi lanes)
            if (mr < M) Out[(size_t)mr * H128 + n] = c[nt][v] + bv;
        }
    }
}

// ---------------------------------------------------------------------------
// Edge-side kernels
// ---------------------------------------------------------------------------
__global__ void zero_int_kernel(int* __restrict__ p, int n)
{
    int i = blockIdx.x * blockDim.x + threadIdx.x;
    if (i < n) p[i] = 0;
}

__global__ void count_edges_kernel(const int* __restrict__ dst,
                                   const int* __restrict__ etype,
                                   int rel, int E, int* __restrict__ cnt)
{
    int e = blockIdx.x * blockDim.x + threadIdx.x;
    if (e < E && etype[e] == rel) atomicAdd(&cnt[dst[e]], 1);
}

// One 32-lane wave per edge; each lane moves 4 features (float4 gather, 4 float atomics).
__global__ void scatter_edges_kernel(const int* __restrict__ src,
                                     const int* __restrict__ dst,
                                     const int* __restrict__ etype,
                                     int rel, int E,
                                     const float* __restrict__ Y,
                                     const int* __restrict__ cnt,
                                     float* __restrict__ Acc)
{
    int g = blockIdx.x * blockDim.x + threadIdx.x;
    int e = g >> 5;
    int lane = g & 31;
    if (e >= E) return;
    if (etype[e] != rel) return;
    int s = src[e];
    int d = dst[e];
    int c = cnt[d]; if (c < 1) c = 1;
    float scale = 1.0f / (float)c;
    v4f y = *(const v4f*)(Y + (size_t)s * H128 + lane * 4);
    float* ap = Acc + (size_t)d * H128 + lane * 4;
    atomicAdd(ap + 0, y.x * scale);
    atomicAdd(ap + 1, y.y * scale);
    atomicAdd(ap + 2, y.z * scale);
    atomicAdd(ap + 3, y.w * scale);
}

__global__ void relu_copy_kernel(const float* __restrict__ in, float* __restrict__ out, int n)
{
    int i = blockIdx.x * blockDim.x + threadIdx.x;
    if (i < n) out[i] = fmaxf(in[i], 0.0f);
}

// Fused relu + linear head + sigmoid: one wave per node, shuffle reduction.
__global__ void head_kernel(const float* __restrict__ Hm, const float* __restrict__ fcw,
                            const float* __restrict__ fcb, float* __restrict__ out, int N)
{
    int g = blockIdx.x * blockDim.x + threadIdx.x;
    int i = g >> 5;
    int lane = g & 31;
    if (i >= N) return;
    v4f h = *(const v4f*)(Hm + (size_t)i * H128 + lane * 4);
    v4f w = *(const v4f*)(fcw + lane * 4);
    float p = fmaxf(h.x, 0.f) * w.x + fmaxf(h.y, 0.f) * w.y +
              fmaxf(h.z, 0.f) * w.z + fmaxf(h.w, 0.f) * w.w;
#pragma unroll
    for (int off = 16; off > 0; off >>= 1) p += __shfl_xor(p, off, 32);
    if (lane == 0) {
        float z = p + fcb[0];
        out[i] = 1.0f / (1.0f + __expf(-z));
    }
}

// ---------------------------------------------------------------------------
// Host-side launch
// ---------------------------------------------------------------------------
extern "C" void kernel_launch(void* const* d_in, const int* in_sizes, int n_in,
                              void* d_out, int out_size, void* d_ws, size_t ws_size,
                              hipStream_t stream)
{
    const float* x     = (const float*)d_in[0];
    const int*   eidx  = (const int*)  d_in[1];
    const int*   etype = (const int*)  d_in[2];
    const float* W1    = (const float*)d_in[3];
    const float* root1 = (const float*)d_in[4];
    const float* b1    = (const float*)d_in[5];
    const float* W2    = (const float*)d_in[6];
    const float* root2 = (const float*)d_in[7];
    const float* b2    = (const float*)d_in[8];
    const float* fcw   = (const float*)d_in[9];
    const float* fcb   = (const float*)d_in[10];
    float* out = (float*)d_out;

    const int N = in_sizes[0] / H128;   // 50000
    const int E = in_sizes[2];          // 600000
    const int* src = eidx;
    const int* dst = eidx + E;

    // Workspace: A (accum) | B (activations) | Y (relation GEMM) | cnt
    float* A = (float*)d_ws;
    float* B = A + (size_t)N * H128;
    float* Y = B + (size_t)N * H128;
    int* cnt = (int*)(Y + (size_t)N * H128);

    const int gemmGrid  = (N + 63) / 64;
    const int nodeGrid  = (N + 255) / 256;
    const int elemGrid  = (N * H128 + 255) / 256;
    const int edgeGrid  = (E + 255) / 256;
    const int edgeWaveG = (E * 32 + 255) / 256;   // 1 wave per edge
    const int headGrid  = (N * 32 + 255) / 256;   // 1 wave per node

    // ---- Layer 1: A = x@root1 + b1 + sum_r mean_r(x@W1r) ----
    gemm128_wmma<<<gemmGrid, 128, 0, stream>>>(x, root1, b1, A, N);
    for (int r = 0; r < 3; ++r) {
        gemm128_wmma<<<gemmGrid, 128, 0, stream>>>(x, W1 + (size_t)r * H128 * H128, nullptr, Y, N);
        zero_int_kernel<<<nodeGrid, 256, 0, stream>>>(cnt, N);
        count_edges_kernel<<<edgeGrid, 256, 0, stream>>>(dst, etype, r, E, cnt);
        scatter_edges_kernel<<<edgeWaveG, 256, 0, stream>>>(src, dst, etype, r, E, Y, cnt, A);
    }
    relu_copy_kernel<<<elemGrid, 256, 0, stream>>>(A, B, N * H128);

    // ---- Layer 2: A = B@root2 + b2 + sum_r mean_r(B@W2r) ----
    gemm128_wmma<<<gemmGrid, 128, 0, stream>>>(B, root2, b2, A, N);
    for (int r = 0; r < 3; ++r) {
        gemm128_wmma<<<gemmGrid, 128, 0, stream>>>(B, W2 + (size_t)r * H128 * H128, nullptr, Y, N);
        zero_int_kernel<<<nodeGrid, 256, 0, stream>>>(cnt, N);
        count_edges_kernel<<<edgeGrid, 256, 0, stream>>>(dst, etype, r, E, cnt);
        scatter_edges_kernel<<<edgeWaveG, 256, 0, stream>>>(src, dst, etype, r, E, Y, cnt, A);
    }

    // ---- Head: sigmoid(relu(A) @ fc_w + fc_b) ----
    head_kernel<<<headGrid, 256, 0, stream>>>(A, fcw, fcb, out, N);
}